// Customized_GatedGraphConv_26912265076898
// MI455X (gfx1250) — compile-verified
//
#include <hip/hip_runtime.h>
#include <math.h>

typedef float v2f __attribute__((ext_vector_type(2)));
typedef float v8f __attribute__((ext_vector_type(8)));

#define N_NODES   30000
#define N_EDGES   600000
#define D_IN      74
#define D_OUT     128
#define F_GATHER  128
#define N_STEPS   6
#define N_ETYPES  5
#define K_CAT     202            // D_OUT + D_IN
#define K_CAT_PAD 208            // padded to multiple of 16 (13*16)

// ---------------------------------------------------------------------------
// h[:, :74] = feat, rest zero
__global__ void k_pad_feat(const float* __restrict__ feat, float* __restrict__ h) {
  int idx = blockIdx.x * blockDim.x + threadIdx.x;
  if (idx >= N_NODES * D_OUT) return;
  int n = idx >> 7, c = idx & 127;
  h[idx] = (c < D_IN) ? feat[n * D_IN + c] : 0.0f;
}

__global__ void k_zero(float* __restrict__ p, int n) {
  int idx = blockIdx.x * blockDim.x + threadIdx.x;
  if (idx < n) p[idx] = 0.0f;
}

// ---------------------------------------------------------------------------
// C[M,N] = A[M,K] @ W[N,K]^T + bias[N]    (torch Linear weight layout)
// fp32 WMMA 16x16x4, LDS-staged tiles.
// Block = 128 threads = 4 waves. Block tile: 128 rows x 64 cols, full K.
// Each wave computes 32 rows x 64 cols = 8 v8f accumulators.
// Dynamic LDS: A-tile [128][K+4] + W-tile [64][K+4] (pad 4 -> conflict-free
// ds_load_b64: 16 rows/half-wave land on disjoint 4-bank groups of 64 banks).
__global__ void k_gemm_lds(const float* __restrict__ A, const float* __restrict__ W,
                           const float* __restrict__ bias, float* __restrict__ C,
                           int M, int N, int K) {
  extern __shared__ float smem[];
  const int KP = K + 4;                         // padded LDS row stride (floats)
  float* As = smem;                             // [128][KP]
  float* Ws = As + 128 * KP;                    // [64][KP]

  const int tid    = threadIdx.x;
  const int mBase0 = blockIdx.x * 128;
  const int nBase  = blockIdx.y * 64;
  const int K4     = K >> 2;

  // ---- stage A tile (zero-fill OOB rows) ----
  for (int f = tid; f < 128 * K4; f += 128) {
    int r = f / K4, c4 = f - r * K4;
    int gr = mBase0 + r;
    float4 v = make_float4(0.f, 0.f, 0.f, 0.f);
    if (gr < M) v = ((const float4*)(A + (size_t)gr * K))[c4];
    *(float4*)&As[r * KP + c4 * 4] = v;
  }
  // ---- stage W tile (zero-fill OOB cols) ----
  for (int f = tid; f < 64 * K4; f += 128) {
    int r = f / K4, c4 = f - r * K4;
    int gc = nBase + r;
    float4 v = make_float4(0.f, 0.f, 0.f, 0.f);
    if (gc < N) v = ((const float4*)(W + (size_t)gc * K))[c4];
    *(float4*)&Ws[r * KP + c4 * 4] = v;
  }
  __syncthreads();

  const int lane = tid & 31;
  const int wave = tid >> 5;                    // 0..3 -> 32-row strip each
  const int half = lane >> 4;                   // K-pair select {0,1} vs {2,3}
  const int l16  = lane & 15;

  const float* aRow0 = &As[(wave * 32 + l16) * KP];
  const float* aRow1 = &As[(wave * 32 + 16 + l16) * KP];

  v8f acc[2][4] = {};

  for (int k = 0; k < K; k += 4) {
    const int kb = k + half * 2;
    v2f a0 = *(const v2f*)(aRow0 + kb);
    v2f a1 = *(const v2f*)(aRow1 + kb);
#pragma unroll
    for (int j = 0; j < 4; ++j) {
      v2f b = *(const v2f*)(&Ws[(j * 16 + l16) * KP + kb]);
      acc[0][j] = __builtin_amdgcn_wmma_f32_16x16x4_f32(
          false, a0, false, b, (short)0, acc[0][j], false, false);
      acc[1][j] = __builtin_amdgcn_wmma_f32_16x16x4_f32(
          false, a1, false, b, (short)0, acc[1][j], false, false);
    }
  }

  // C/D layout: vgpr i -> row + i + half*8 ; lane%16 -> column
#pragma unroll
  for (int s = 0; s < 2; ++s) {
#pragma unroll
    for (int j = 0; j < 4; ++j) {
      int col = nBase + j * 16 + l16;
      if (col < N) {
        float bv = bias[col];
#pragma unroll
        for (int i = 0; i < 8; ++i) {
          int row = mBase0 + wave * 32 + s * 16 + i + half * 8;
          if (row < M) C[(size_t)row * N + col] = acc[s][j][i] + bv;
        }
      }
    }
  }
}

// ---------------------------------------------------------------------------
// one wave per edge: a[dst] += all_proj[src, etype*128 : etype*128+128]
__global__ void k_edge_scatter(const float* __restrict__ all_proj,
                               const int* __restrict__ src, const int* __restrict__ dst,
                               const int* __restrict__ ety, float* __restrict__ agg) {
  int wave = (blockIdx.x * blockDim.x + threadIdx.x) >> 5;
  int lane = threadIdx.x & 31;
  if (wave >= N_EDGES) return;
  int s = src[wave], d = dst[wave], t = ety[wave];
  const float4* row = (const float4*)(all_proj + (size_t)s * (N_ETYPES * D_OUT) + (size_t)t * D_OUT);
  float4 v = row[lane];
  float* out = agg + (size_t)d * D_OUT + lane * 4;
  __hip_atomic_fetch_add(out + 0, v.x, __ATOMIC_RELAXED, __HIP_MEMORY_SCOPE_AGENT);
  __hip_atomic_fetch_add(out + 1, v.y, __ATOMIC_RELAXED, __HIP_MEMORY_SCOPE_AGENT);
  __hip_atomic_fetch_add(out + 2, v.z, __ATOMIC_RELAXED, __HIP_MEMORY_SCOPE_AGENT);
  __hip_atomic_fetch_add(out + 3, v.w, __ATOMIC_RELAXED, __HIP_MEMORY_SCOPE_AGENT);
}

// ---------------------------------------------------------------------------
// GRU elementwise: h = (1-z)*n + z*h   (gate order r,z,n; biases already in gi/gh)
__global__ void k_gru(const float* __restrict__ gi, const float* __restrict__ gh,
                      float* __restrict__ h) {
  int idx = blockIdx.x * blockDim.x + threadIdx.x;
  if (idx >= N_NODES * D_OUT) return;
  int n = idx >> 7, c = idx & 127;
  const float* gin = gi + (size_t)n * (3 * D_OUT);
  const float* ghn = gh + (size_t)n * (3 * D_OUT);
  float r  = 1.0f / (1.0f + __expf(-(gin[c] + ghn[c])));
  float z  = 1.0f / (1.0f + __expf(-(gin[D_OUT + c] + ghn[D_OUT + c])));
  float nn = tanhf(gin[2 * D_OUT + c] + r * ghn[2 * D_OUT + c]);
  h[idx] = (1.0f - z) * nn + z * h[idx];
}

// ---------------------------------------------------------------------------
// cat = [h | feat | 0-pad] padded to K_CAT_PAD columns
__global__ void k_build_cat(const float* __restrict__ h, const float* __restrict__ feat,
                            float* __restrict__ cat) {
  int idx = blockIdx.x * blockDim.x + threadIdx.x;
  if (idx >= N_NODES * K_CAT_PAD) return;
  int n = idx / K_CAT_PAD, c = idx % K_CAT_PAD;
  float v = 0.0f;
  if (c < D_OUT)      v = h[(size_t)n * D_OUT + c];
  else if (c < K_CAT) v = feat[(size_t)n * D_IN + (c - D_OUT)];
  cat[idx] = v;
}

// zero-pad i_W [128 x 202] -> [128 x 208]
__global__ void k_pad_iw(const float* __restrict__ iW, float* __restrict__ iWp) {
  int idx = blockIdx.x * blockDim.x + threadIdx.x;
  if (idx >= F_GATHER * K_CAT_PAD) return;
  int r = idx / K_CAT_PAD, c = idx % K_CAT_PAD;
  iWp[idx] = (c < K_CAT) ? iW[(size_t)r * K_CAT + c] : 0.0f;
}

// out = sigmoid(gatePre) * jout
__global__ void k_final(const float* __restrict__ gatePre, const float* __restrict__ jout,
                        float* __restrict__ out) {
  int idx = blockIdx.x * blockDim.x + threadIdx.x;
  if (idx >= N_NODES * F_GATHER) return;
  float g = 1.0f / (1.0f + __expf(-gatePre[idx]));
  out[idx] = g * jout[idx];
}

// ---------------------------------------------------------------------------
extern "C" void kernel_launch(void* const* d_in, const int* in_sizes, int n_in,
                              void* d_out, int out_size, void* d_ws, size_t ws_size,
                              hipStream_t stream) {
  (void)in_sizes; (void)n_in; (void)out_size; (void)ws_size;

  const float* feat  = (const float*)d_in[0];
  const int*   src   = (const int*)  d_in[1];
  const int*   dst   = (const int*)  d_in[2];
  const int*   ety   = (const int*)  d_in[3];
  const float* lin_W = (const float*)d_in[4];   // [5,128,128] -> flat [640,128]
  const float* lin_b = (const float*)d_in[5];   // flat [640]
  const float* W_ih  = (const float*)d_in[6];   // [384,128]
  const float* W_hh  = (const float*)d_in[7];   // [384,128]
  const float* b_ih  = (const float*)d_in[8];
  const float* b_hh  = (const float*)d_in[9];
  const float* i_W   = (const float*)d_in[10];  // [128,202]
  const float* i_b   = (const float*)d_in[11];
  const float* j_W   = (const float*)d_in[12];  // [128,128]
  const float* j_b   = (const float*)d_in[13];
  float* out = (float*)d_out;

  // workspace layout (floats)
  float* ws   = (float*)d_ws;
  float* h    = ws;                                      // N*128
  float* proj = h    + (size_t)N_NODES * D_OUT;          // N*640 (reused: cat N*208)
  float* agg  = proj + (size_t)N_NODES * N_ETYPES*D_OUT; // N*128 (reused: padded i_W)
  float* gi   = agg  + (size_t)N_NODES * D_OUT;          // N*384 (reused: gate pre-act)
  float* gh   = gi   + (size_t)N_NODES * 3 * D_OUT;      // N*384 (reused: j-output)

  const int TB = 256;
  const int nElemH = N_NODES * D_OUT;
  dim3 blkGemm(128);
  const int gmx = (N_NODES + 127) / 128;                 // 128 rows per block

  const size_t lds128 = (size_t)(128 + 64) * (D_OUT + 4) * sizeof(float);     // K=128
  const size_t lds208 = (size_t)(128 + 64) * (K_CAT_PAD + 4) * sizeof(float); // K=208

  // h = pad(feat)
  k_pad_feat<<<(nElemH + TB - 1) / TB, TB, 0, stream>>>(feat, h);

  for (int step = 0; step < N_STEPS; ++step) {
    // all_proj = h @ lin_W_flat^T + lin_b      [N x 640]
    k_gemm_lds<<<dim3(gmx, (N_ETYPES * D_OUT) / 64), blkGemm, lds128, stream>>>(
        h, lin_W, lin_b, proj, N_NODES, N_ETYPES * D_OUT, D_OUT);

    // a = segment_sum over edges
    k_zero<<<(nElemH + TB - 1) / TB, TB, 0, stream>>>(agg, nElemH);
    k_edge_scatter<<<(N_EDGES * 32 + TB - 1) / TB, TB, 0, stream>>>(proj, src, dst, ety, agg);

    // gi = a @ W_ih^T + b_ih ; gh = h @ W_hh^T + b_hh   [N x 384]
    k_gemm_lds<<<dim3(gmx, (3 * D_OUT) / 64), blkGemm, lds128, stream>>>(
        agg, W_ih, b_ih, gi, N_NODES, 3 * D_OUT, D_OUT);
    k_gemm_lds<<<dim3(gmx, (3 * D_OUT) / 64), blkGemm, lds128, stream>>>(
        h, W_hh, b_hh, gh, N_NODES, 3 * D_OUT, D_OUT);

    // h = GRU(a, h)
    k_gru<<<(nElemH + TB - 1) / TB, TB, 0, stream>>>(gi, gh, h);
  }

  // final gated combine
  float* cat     = proj;   // N x 208
  float* iWp     = agg;    // 128 x 208
  float* gatePre = gi;     // N x 128
  float* jout    = gh;     // N x 128

  k_build_cat<<<(N_NODES * K_CAT_PAD + TB - 1) / TB, TB, 0, stream>>>(h, feat, cat);
  k_pad_iw<<<(F_GATHER * K_CAT_PAD + TB - 1) / TB, TB, 0, stream>>>(i_W, iWp);

  k_gemm_lds<<<dim3(gmx, F_GATHER / 64), blkGemm, lds208, stream>>>(
      cat, iWp, i_b, gatePre, N_NODES, F_GATHER, K_CAT_PAD);
  k_gemm_lds<<<dim3(gmx, F_GATHER / 64), blkGemm, lds128, stream>>>(
      h, j_W, j_b, jout, N_NODES, F_GATHER, D_OUT);

  k_final<<<(N_NODES * F_GATHER + TB - 1) / TB, TB, 0, stream>>>(gatePre, jout, out);
}